// MultiHeadAttention_3332894622122
// MI455X (gfx1250) — compile-verified
//
#include <hip/hip_runtime.h>
#include <hip/hip_bf16.h>
#include <math.h>

typedef float v2f __attribute__((ext_vector_type(2)));
typedef float v4f __attribute__((ext_vector_type(4)));
typedef float v8f __attribute__((ext_vector_type(8)));

#define D_MODEL 1024
#define N_HEAD  16
#define D_K     64
#define BATCH   2
#define SEQ     2048
#define MROWS   (BATCH * SEQ)   // 4096
#define INV_SCALE 0.125f        // 1/sqrt(64)

// LDS strides in floats: multiples of 4 (16B) for async B128 stores,
// 36 mod 64 = 36 and 68 mod 64 = 4 keep 16-lane column reads conflict-free.
#define STRA 36
#define STRB 68

__device__ __forceinline__ v8f wmma4(v2f a, v2f b, v8f c) {
  // D = A(16x4,f32) * B(4x16,f32) + C(16x16,f32), wave32
  return __builtin_amdgcn_wmma_f32_16x16x4_f32(
      /*neg_a=*/false, a, /*neg_b=*/false, b,
      /*c_mod=*/(short)0, c, /*reuse_a=*/false, /*reuse_b=*/false);
}

// Async DMA: memory -> LDS, 16 bytes per lane, GVS addressing
// (SGPR 64-bit base + per-lane 32-bit byte offset). Tracked by ASYNCcnt.
__device__ __forceinline__ void async_copy_b128(void* lds_dst, const void* gbase,
                                                unsigned byte_off) {
  unsigned lds = (unsigned)(size_t)lds_dst;  // low 32 bits = LDS address
  asm volatile("global_load_async_to_lds_b128 %0, %1, %2"
               :
               : "v"(lds), "v"(byte_off), "s"(gbase)
               : "memory");
}

__device__ __forceinline__ void async_wait0() {
  asm volatile("s_wait_asynccnt 0x0" ::: "memory");
}

// -----------------------------------------------------------------------------
// C[M x N] = A[M x 1024] @ W[1024 x N] + bias[N];  M=4096, N=1024
// Block: 256 threads (8 waves). Block tile 128x64; waves 4(m) x 2(n),
// each wave computes 32x32 (4 WMMA accumulators, 1 LDS frag-load per WMMA).
// Double-buffered async LDS staging (DMA overlaps WMMA).
// -----------------------------------------------------------------------------
__device__ __forceinline__ void stage_gemm_tiles(float* AsBuf, float* WsBuf,
                                                 const float* A, const float* W,
                                                 int m0b, int n0b, int k0, int t) {
  // A tile: 128 x 32 (4 async B128 per thread)
  #pragma unroll
  for (int i = 0; i < 4; ++i) {
    int idx = i * 256 + t;
    int r = idx >> 3, c = (idx & 7) << 2;
    async_copy_b128(&AsBuf[r * STRA + c], A,
                    (unsigned)(((m0b + r) * D_MODEL + k0 + c) * 4));
  }
  // W tile: 32 x 64 (2 async B128 per thread)
  #pragma unroll
  for (int i = 0; i < 2; ++i) {
    int idx = i * 256 + t;
    int r = idx >> 4, c = (idx & 15) << 2;
    async_copy_b128(&WsBuf[r * STRB + c], W,
                    (unsigned)(((k0 + r) * D_MODEL + n0b + c) * 4));
  }
}

__global__ __launch_bounds__(256)
void gemm_bias_wmma(const float* __restrict__ A, const float* __restrict__ W,
                    const float* __restrict__ bias, float* __restrict__ C) {
  const int K = D_MODEL, N = D_MODEL;
  __shared__ float As[2][128 * STRA];
  __shared__ float Ws[2][32 * STRB];

  const int t    = threadIdx.x;
  const int lane = t & 31;
  const int wid  = t >> 5;
  const int m0b  = blockIdx.y * 128;
  const int n0b  = blockIdx.x * 64;

  const int ml = lane & 15;
  const int hi = lane >> 4;           // 0: K+{0,1}, 1: K+{2,3}
  const int m0 = (wid >> 1) * 32;     // wave row base within block tile
  const int n0 = (wid & 1) * 32;      // wave col base within block tile

  v8f c00 = {}, c01 = {}, c10 = {}, c11 = {};

  stage_gemm_tiles(As[0], Ws[0], A, W, m0b, n0b, 0, t);
  async_wait0();
  __syncthreads();

  for (int k0 = 0; k0 < K; k0 += 32) {
    const int cur = (k0 >> 5) & 1;
    const float* Acur = As[cur];
    const float* Wcur = Ws[cur];
    if (k0 + 32 < K)
      stage_gemm_tiles(As[cur ^ 1], Ws[cur ^ 1], A, W, m0b, n0b, k0 + 32, t);

    #pragma unroll
    for (int kk = 0; kk < 32; kk += 4) {
      int kb = kk + hi * 2;
      v2f a0; a0.x = Acur[(m0 + ml) * STRA + kb];      a0.y = Acur[(m0 + ml) * STRA + kb + 1];
      v2f a1; a1.x = Acur[(m0 + 16 + ml) * STRA + kb]; a1.y = Acur[(m0 + 16 + ml) * STRA + kb + 1];
      v2f b0; b0.x = Wcur[kb * STRB + n0 + ml];        b0.y = Wcur[(kb + 1) * STRB + n0 + ml];
      v2f b1; b1.x = Wcur[kb * STRB + n0 + 16 + ml];   b1.y = Wcur[(kb + 1) * STRB + n0 + 16 + ml];
      c00 = wmma4(a0, b0, c00);
      c01 = wmma4(a0, b1, c01);
      c10 = wmma4(a1, b0, c10);
      c11 = wmma4(a1, b1, c11);
    }
    async_wait0();
    __syncthreads();
  }

  #pragma unroll
  for (int i = 0; i < 8; ++i) {
    int row = m0b + m0 + i + hi * 8;
    int col = n0b + n0 + ml;
    float bi0 = bias[col], bi1 = bias[col + 16];
    C[(size_t)row * N + col]             = c00[i] + bi0;
    C[(size_t)row * N + col + 16]        = c01[i] + bi1;
    C[(size_t)(row + 16) * N + col]      = c10[i] + bi0;
    C[(size_t)(row + 16) * N + col + 16] = c11[i] + bi1;
  }
}

// -----------------------------------------------------------------------------
// Raw scores: scores[b,h,q,k] = dot(Qh[q,:], Kh[k,:]) / sqrt(dk)
// Block tile: 32 q-rows x 128 k-cols per (b,h). 8 waves as 2x4, wave: 16x32.
// Single-stage (K = 64 only), async LDS staging.
// -----------------------------------------------------------------------------
__global__ __launch_bounds__(256)
void attn_scores_wmma(const float* __restrict__ Qp, const float* __restrict__ Kp,
                      float* __restrict__ attn) {
  __shared__ float Qs[32 * STRB];
  __shared__ float Ks[128 * STRB];

  const int t    = threadIdx.x;
  const int lane = t & 31;
  const int wid  = t >> 5;
  const int bh   = blockIdx.z;
  const int b    = bh >> 4;
  const int h    = bh & 15;
  const int q0   = blockIdx.y * 32;
  const int k0   = blockIdx.x * 128;

  const float* Qbase = Qp + (size_t)b * SEQ * D_MODEL + h * D_K;
  const float* Kbase = Kp + (size_t)b * SEQ * D_MODEL + h * D_K;

  // Stage Q tile: 32 x 64
  #pragma unroll
  for (int i = 0; i < 2; ++i) {
    int idx = i * 256 + t;
    int r = idx >> 4, c = (idx & 15) << 2;
    async_copy_b128(&Qs[r * STRB + c], Qbase,
                    (unsigned)(((q0 + r) * D_MODEL + c) * 4));
  }
  // Stage K tile: 128 x 64
  #pragma unroll
  for (int i = 0; i < 8; ++i) {
    int idx = i * 256 + t;
    int r = idx >> 4, c = (idx & 15) << 2;
    async_copy_b128(&Ks[r * STRB + c], Kbase,
                    (unsigned)(((k0 + r) * D_MODEL + c) * 4));
  }
  async_wait0();
  __syncthreads();

  const int wr = wid >> 2, wc = wid & 3;
  const int m0 = wr * 16, n0 = wc * 32;
  const int ml = lane & 15, hi = lane >> 4;

  v8f c0 = {}; v8f c1 = {};
  #pragma unroll
  for (int kk = 0; kk < 64; kk += 4) {
    int kb = kk + hi * 2;
    v2f a;  a.x  = Qs[(m0 + ml) * STRB + kb];       a.y  = Qs[(m0 + ml) * STRB + kb + 1];
    // B = K^T: B[d][n] = K[n][d]
    v2f b0; b0.x = Ks[(n0 + ml) * STRB + kb];       b0.y = Ks[(n0 + ml) * STRB + kb + 1];
    v2f b1; b1.x = Ks[(n0 + 16 + ml) * STRB + kb];  b1.y = Ks[(n0 + 16 + ml) * STRB + kb + 1];
    c0 = wmma4(a, b0, c0);
    c1 = wmma4(a, b1, c1);
  }

  #pragma unroll
  for (int i = 0; i < 8; ++i) {
    int qq = q0 + m0 + i + hi * 8;
    int kc = k0 + n0 + ml;
    size_t base = ((size_t)bh * SEQ + qq) * SEQ;
    attn[base + kc]      = c0[i] * INV_SCALE;
    attn[base + kc + 16] = c1[i] * INV_SCALE;
  }
}

// -----------------------------------------------------------------------------
// In-place row softmax over the 2048-wide score rows.
// -----------------------------------------------------------------------------
__global__ __launch_bounds__(256)
void softmax_rows(float* __restrict__ attn) {
  __shared__ float red[256];
  const int t  = threadIdx.x;
  const int q  = blockIdx.x;
  const int bh = blockIdx.y;
  float* p = attn + ((size_t)bh * SEQ + q) * SEQ;

  float vals[8];
  float m = -INFINITY;
  #pragma unroll
  for (int j = 0; j < 8; ++j) { vals[j] = p[t + j * 256]; m = fmaxf(m, vals[j]); }
  red[t] = m; __syncthreads();
  for (int s = 128; s > 0; s >>= 1) {
    if (t < s) red[t] = fmaxf(red[t], red[t + s]);
    __syncthreads();
  }
  m = red[0]; __syncthreads();

  float sum = 0.f;
  #pragma unroll
  for (int j = 0; j < 8; ++j) { vals[j] = __expf(vals[j] - m); sum += vals[j]; }
  red[t] = sum; __syncthreads();
  for (int s = 128; s > 0; s >>= 1) {
    if (t < s) red[t] += red[t + s];
    __syncthreads();
  }
  float inv = 1.0f / red[0];
  #pragma unroll
  for (int j = 0; j < 8; ++j) p[t + j * 256] = vals[j] * inv;
}

// -----------------------------------------------------------------------------
// ctx[b,q,h*64+d] = sum_k attn[b,h,q,k] * Vp[b,k,h*64+d]
// Per (b,h): M=2048, N=64, K=2048. Block tile 64 x 64; 8 waves as 2(m) x 4(n),
// each wave 32x16 (2 accumulators). Double-buffered async staging.
// -----------------------------------------------------------------------------
__device__ __forceinline__ void stage_ctx_tiles(float* AsBuf, float* BsBuf,
                                                const float* Arow, const float* Vbase,
                                                int k0, int t) {
  // attn tile: 64 x 32 (2 async B128 per thread)
  #pragma unroll
  for (int i = 0; i < 2; ++i) {
    int idx = i * 256 + t;
    int r = idx >> 3, c = (idx & 7) << 2;
    async_copy_b128(&AsBuf[r * STRA + c], Arow,
                    (unsigned)((r * SEQ + k0 + c) * 4));
  }
  // V tile: 32 x 64 (2 async B128 per thread)
  #pragma unroll
  for (int i = 0; i < 2; ++i) {
    int idx = i * 256 + t;
    int r = idx >> 4, c = (idx & 15) << 2;
    async_copy_b128(&BsBuf[r * STRB + c], Vbase,
                    (unsigned)(((k0 + r) * D_MODEL + c) * 4));
  }
}

__global__ __launch_bounds__(256)
void attn_ctx_wmma(const float* __restrict__ attn, const float* __restrict__ Vp,
                   float* __restrict__ ctx) {
  __shared__ float As[2][64 * STRA];
  __shared__ float Bs[2][32 * STRB];

  const int t    = threadIdx.x;
  const int lane = t & 31;
  const int wid  = t >> 5;
  const int bh   = blockIdx.y;
  const int b    = bh >> 4;
  const int h    = bh & 15;
  const int q0   = blockIdx.x * 64;

  const float* Arow  = attn + ((size_t)bh * SEQ + q0) * SEQ;
  const float* Vbase = Vp + (size_t)b * SEQ * D_MODEL + h * D_K;

  const int wr = wid >> 2, wc = wid & 3;
  const int m0 = wr * 32, n0 = wc * 16;
  const int ml = lane & 15, hi = lane >> 4;

  v8f acc0 = {}, acc1 = {};

  stage_ctx_tiles(As[0], Bs[0], Arow, Vbase, 0, t);
  async_wait0();
  __syncthreads();

  for (int k0 = 0; k0 < SEQ; k0 += 32) {
    const int cur = (k0 >> 5) & 1;
    const float* Acur = As[cur];
    const float* Bcur = Bs[cur];
    if (k0 + 32 < SEQ)
      stage_ctx_tiles(As[cur ^ 1], Bs[cur ^ 1], Arow, Vbase, k0 + 32, t);

    #pragma unroll
    for (int kk = 0; kk < 32; kk += 4) {
      int kb = kk + hi * 2;
      v2f a0; a0.x = Acur[(m0 + ml) * STRA + kb];      a0.y = Acur[(m0 + ml) * STRA + kb + 1];
      v2f a1; a1.x = Acur[(m0 + 16 + ml) * STRA + kb]; a1.y = Acur[(m0 + 16 + ml) * STRA + kb + 1];
      v2f bb; bb.x = Bcur[kb * STRB + n0 + ml];        bb.y = Bcur[(kb + 1) * STRB + n0 + ml];
      acc0 = wmma4(a0, bb, acc0);
      acc1 = wmma4(a1, bb, acc1);
    }
    async_wait0();
    __syncthreads();
  }

  #pragma unroll
  for (int i = 0; i < 8; ++i) {
    int qq  = q0 + m0 + i + hi * 8;
    int col = h * D_K + n0 + ml;
    ctx[((size_t)b * SEQ + qq) * D_MODEL + col]      = acc0[i];
    ctx[((size_t)b * SEQ + qq + 16) * D_MODEL + col] = acc1[i];
  }
}

// -----------------------------------------------------------------------------
extern "C" void kernel_launch(void* const* d_in, const int* in_sizes, int n_in,
                              void* d_out, int out_size, void* d_ws, size_t ws_size,
                              hipStream_t stream) {
  const float* q  = (const float*)d_in[0];
  const float* k  = (const float*)d_in[1];
  const float* v  = (const float*)d_in[2];
  const float* Wq = (const float*)d_in[3];
  const float* bq = (const float*)d_in[4];
  const float* Wk = (const float*)d_in[5];
  const float* bk = (const float*)d_in[6];
  const float* Wv = (const float*)d_in[7];
  const float* bv = (const float*)d_in[8];
  const float* Wo = (const float*)d_in[9];
  const float* bo = (const float*)d_in[10];

  float* out  = (float*)d_out;                          // [B, S, D_MODEL]
  float* attn = out + (size_t)MROWS * D_MODEL;          // [B, H, S, S]

  float* Qp  = (float*)d_ws;                            // [B*S, D_MODEL]
  float* Kp  = Qp + (size_t)MROWS * D_MODEL;
  float* Vp  = Kp + (size_t)MROWS * D_MODEL;
  float* ctx = Vp + (size_t)MROWS * D_MODEL;

  dim3 blk(256);
  dim3 gGemm(D_MODEL / 64, MROWS / 128);
  gemm_bias_wmma<<<gGemm, blk, 0, stream>>>(q, Wq, bq, Qp);
  gemm_bias_wmma<<<gGemm, blk, 0, stream>>>(k, Wk, bk, Kp);
  gemm_bias_wmma<<<gGemm, blk, 0, stream>>>(v, Wv, bv, Vp);

  dim3 gScore(SEQ / 128, SEQ / 32, BATCH * N_HEAD);
  attn_scores_wmma<<<gScore, blk, 0, stream>>>(Qp, Kp, attn);

  dim3 gSm(SEQ, BATCH * N_HEAD);
  softmax_rows<<<gSm, blk, 0, stream>>>(attn);

  dim3 gCtx(SEQ / 64, BATCH * N_HEAD);
  attn_ctx_wmma<<<gCtx, blk, 0, stream>>>(attn, Vp, ctx);

  gemm_bias_wmma<<<gGemm, blk, 0, stream>>>(ctx, Wo, bo, out);
}